// LowRankBilinearAttention_20882130993382
// MI455X (gfx1250) — compile-verified
//
#include <hip/hip_runtime.h>
#include <hip/hip_bf16.h>
#include <math.h>

#define B_   16
#define P_   196
#define L_   80
#define D1_  2048
#define D2_  300
#define ATT_ 512

typedef float v2f __attribute__((ext_vector_type(2)));
typedef float v8f __attribute__((ext_vector_type(8)));

#if __has_builtin(__builtin_amdgcn_tanhf)
#define TANHF(x) __builtin_amdgcn_tanhf(x)
#else
#define TANHF(x) tanhf(x)
#endif

// ---------------------------------------------------------------------------
// v[a] = sum_c Wt[c] * Wh[c,a]   (folds the 131 GFLOP blpa,ca GEMM away)
// c0   = sum_c bh[c] * Wt[c] + bt
// ---------------------------------------------------------------------------
__global__ void __launch_bounds__(ATT_) prep_v_kernel(
    const float* __restrict__ Wh, const float* __restrict__ bh,
    const float* __restrict__ Wt, const float* __restrict__ bt,
    float* __restrict__ v, float* __restrict__ c0)
{
    const int a = threadIdx.x;
    float acc = 0.f;
    for (int c = 0; c < ATT_; ++c)
        acc += Wt[c] * Wh[(size_t)c * ATT_ + a];   // coalesced across threads
    v[a] = acc;
    if (a == 0) {
        float s = 0.f;
        for (int c = 0; c < ATT_; ++c) s += bh[c] * Wt[c];
        *c0 = s + bt[0];
    }
}

// ---------------------------------------------------------------------------
// C[M,N] = A[M,K] * W[N,K]^T  — fp32 WMMA 16x16x4, one wave per 16x16 C tile.
// A lane layout (§7.12.2 f32 A 16x4): lane m=lane&15, a[j]=A[m][2*(lane>>4)+j]
// B lane layout (f32 B 4x16):        lane n=lane&15, b[j]=B[2*(lane>>4)+j][n]
//   with B[k][n] = W[(col0+n)*ldw + k]  (weight stored (out,in) like torch)
// C layout: VGPR r -> row r + 8*(lane>>4), col = lane&15
// ---------------------------------------------------------------------------
__global__ void __launch_bounds__(32) gemm_f32_wt(
    const float* __restrict__ A, const float* __restrict__ W,
    float* __restrict__ C, int K, int lda, int ldw, int ldc)
{
    const int lane = threadIdx.x;
    const int half = lane >> 4;
    const int mn   = lane & 15;
    const int row0 = blockIdx.y << 4;
    const int col0 = blockIdx.x << 4;

    const float* __restrict__ arow = A + (size_t)(row0 + mn) * lda;
    const float* __restrict__ wrow = W + (size_t)(col0 + mn) * ldw;

    v8f acc = {};
    for (int k0 = 0; k0 < K; k0 += 4) {
        const int k = k0 + 2 * half;
        v2f a; a.x = arow[k]; a.y = arow[k + 1];     // global_load_b64
        v2f b; b.x = wrow[k]; b.y = wrow[k + 1];     // global_load_b64
        acc = __builtin_amdgcn_wmma_f32_16x16x4_f32(
            false, a, false, b, (short)0, acc, false, false);
    }

    float* __restrict__ crow = C + (size_t)(row0 + 8 * half) * ldc + col0 + mn;
#pragma unroll
    for (int r = 0; r < 8; ++r)
        crow[(size_t)r * ldc] = acc[r];
}

// ---------------------------------------------------------------------------
// Batched C[z] = A[z][M,K] * Bm[z][K,N]  (B in natural K-major orientation)
// Used for label = alpha · x1 (per-batch 80 x 196 x 2048).
// ---------------------------------------------------------------------------
__global__ void __launch_bounds__(32) gemm_f32_bn(
    const float* __restrict__ A, const float* __restrict__ Bm,
    float* __restrict__ C, int K, int lda, int ldb, int ldc,
    long strA, long strB, long strC)
{
    const int lane = threadIdx.x;
    const int half = lane >> 4;
    const int mn   = lane & 15;
    const int row0 = blockIdx.y << 4;
    const int col0 = blockIdx.x << 4;

    const float* __restrict__ arow = A + (size_t)blockIdx.z * strA + (size_t)(row0 + mn) * lda;
    const float* __restrict__ bcol = Bm + (size_t)blockIdx.z * strB + col0 + mn;

    v8f acc = {};
    for (int k0 = 0; k0 < K; k0 += 4) {
        const int k = k0 + 2 * half;
        v2f a; a.x = arow[k];                 a.y = arow[k + 1];
        v2f b; b.x = bcol[(size_t)k * ldb];   b.y = bcol[(size_t)(k + 1) * ldb];
        acc = __builtin_amdgcn_wmma_f32_16x16x4_f32(
            false, a, false, b, (short)0, acc, false, false);
    }

    float* __restrict__ crow = C + (size_t)blockIdx.z * strC
                                 + (size_t)(row0 + 8 * half) * ldc + col0 + mn;
#pragma unroll
    for (int r = 0; r < 8; ++r)
        crow[(size_t)r * ldc] = acc[r];
}

// ---------------------------------------------------------------------------
// Fused score + softmax: one block per (b,l).
//   t[p] = sum_a tanh(p1[b,p,a] * p2[b,l,a]) * v[a] + c0
//   alpha[b,l,:] = softmax_p(t)
// p2 row + v staged in LDS; 8 waves each own strided p's; wave32 shfl reduce.
// Transcendental (tanh/exp) bound; p1/p2 are L2-resident (9 MB << 192 MB).
// ---------------------------------------------------------------------------
__global__ void __launch_bounds__(256) score_softmax_kernel(
    const float* __restrict__ p1, const float* __restrict__ p2,
    const float* __restrict__ v, const float* __restrict__ c0p,
    float* __restrict__ alpha)
{
    __shared__ float p2s[ATT_];
    __shared__ float vs[ATT_];
    __shared__ float ts[P_];
    __shared__ float red[256];

    const int tid = threadIdx.x;
    const int bl  = blockIdx.x;          // b*L + l
    const int b   = bl / L_;

    const float* __restrict__ p2row = p2 + (size_t)bl * ATT_;
    for (int i = tid; i < ATT_; i += 256) { p2s[i] = p2row[i]; vs[i] = v[i]; }
    __syncthreads();

    const float c0   = *c0p;
    const int   wave = tid >> 5;
    const int   lane = tid & 31;

    for (int p = wave; p < P_; p += 8) {
        const float* __restrict__ p1row = p1 + ((size_t)b * P_ + p) * ATT_;
        if (p + 8 < P_)
            __builtin_prefetch(p1 + ((size_t)b * P_ + p + 8) * ATT_ + lane * 4, 0, 0);
        float acc = 0.f;
        for (int a = lane; a < ATT_; a += 32)
            acc += TANHF(p1row[a] * p2s[a]) * vs[a];
#pragma unroll
        for (int off = 16; off > 0; off >>= 1)
            acc += __shfl_xor(acc, off, 32);
        if (lane == 0) ts[p] = acc + c0;             // TAU == 1
    }
    __syncthreads();

    // softmax over P_ = 196
    float mx = -3.402823466e38f;
    for (int p = tid; p < P_; p += 256) mx = fmaxf(mx, ts[p]);
    red[tid] = mx; __syncthreads();
    for (int s = 128; s > 0; s >>= 1) {
        if (tid < s) red[tid] = fmaxf(red[tid], red[tid + s]);
        __syncthreads();
    }
    mx = red[0]; __syncthreads();

    float sum = 0.f;
    for (int p = tid; p < P_; p += 256) { float e = __expf(ts[p] - mx); ts[p] = e; sum += e; }
    red[tid] = sum; __syncthreads();
    for (int s = 128; s > 0; s >>= 1) {
        if (tid < s) red[tid] += red[tid + s];
        __syncthreads();
    }
    const float inv = 1.f / red[0];

    float* __restrict__ arow = alpha + (size_t)bl * P_;
    for (int p = tid; p < P_; p += 256) arow[p] = ts[p] * inv;
}

// ---------------------------------------------------------------------------
extern "C" void kernel_launch(void* const* d_in, const int* in_sizes, int n_in,
                              void* d_out, int out_size, void* d_ws, size_t ws_size,
                              hipStream_t stream)
{
    const float* x1 = (const float*)d_in[0];
    const float* x2 = (const float*)d_in[1];
    const float* W1 = (const float*)d_in[2];
    const float* W2 = (const float*)d_in[3];
    const float* Wh = (const float*)d_in[4];
    const float* bh = (const float*)d_in[5];
    const float* Wt = (const float*)d_in[6];
    const float* bt = (const float*)d_in[7];

    float* out   = (float*)d_out;
    float* label = out;                                  // (B,L,D1)
    float* alpha = out + (size_t)B_ * L_ * D1_;          // (B,L,P)

    float* ws = (float*)d_ws;
    float* v  = ws;                                      // 512
    float* c0 = ws + ATT_;                               // 1
    float* p1 = ws + 2 * ATT_;                           // B*P*ATT
    float* p2 = p1 + (size_t)B_ * P_ * ATT_;             // B*L*ATT

    // 1) fold Wh/wt into a single 512-vector
    hipLaunchKernelGGL(prep_v_kernel, dim3(1), dim3(ATT_), 0, stream,
                       Wh, bh, Wt, bt, v, c0);

    // 2) p1 = x1 · W1^T   : M = B*P = 3136, K = 2048, N = 512   (fp32 WMMA)
    hipLaunchKernelGGL(gemm_f32_wt, dim3(ATT_ / 16, (B_ * P_) / 16), dim3(32), 0, stream,
                       x1, W1, p1, D1_, D1_, D1_, ATT_);

    // 3) p2 = x2 · W2^T   : M = B*L = 1280, K = 300, N = 512    (fp32 WMMA)
    hipLaunchKernelGGL(gemm_f32_wt, dim3(ATT_ / 16, (B_ * L_) / 16), dim3(32), 0, stream,
                       x2, W2, p2, D2_, D2_, D2_, ATT_);

    // 4) fused tanh-bilinear score + softmax -> alpha
    hipLaunchKernelGGL(score_softmax_kernel, dim3(B_ * L_), dim3(256), 0, stream,
                       p1, p2, v, c0, alpha);

    // 5) label = alpha · x1 : per batch 80 x 196 x 2048          (fp32 WMMA)
    hipLaunchKernelGGL(gemm_f32_bn, dim3(D1_ / 16, L_ / 16, B_), dim3(32), 0, stream,
                       alpha, x1, label, P_, P_, D1_, D1_,
                       (long)L_ * P_, (long)P_ * D1_, (long)L_ * D1_);
}